// RGCN_40080634806758
// MI455X (gfx1250) — compile-verified
//
#include <hip/hip_runtime.h>
#include <hip/hip_bf16.h>
#include <math.h>

// RGCN on MI455X (gfx1250).
// Memory-bound workload (~1.1 GB of edge gather/scatter vs only ~26 GFLOP of GEMM),
// so we keep full fp32 precision and use V_WMMA_F32_16X16X4_F32 for all dense math.
// W (64KB) is staged TRANSPOSED in LDS per workgroup (8 waves share it): each WMMA
// B operand {W[k][c], W[k+1][c]} becomes one aligned ds_load_b64, and with column
// stride 132 (== 4 mod 64) the two half-wave access patterns hit disjoint LDS banks.

#define RREL 8
#define HIDD 128
#define NCLS 16
#define NGRAPH 64
#define WSTR 132   // LDS column stride in words; 132 mod 64 == 4 -> conflict-free b64 reads

typedef float v2f __attribute__((ext_vector_type(2)));
typedef float v8f __attribute__((ext_vector_type(8)));

__device__ __forceinline__ void atomic_add_f32(float* p, float v) {
    // lowers to global_atomic_add_f32 (agent scope), no CAS loop
    __hip_atomic_fetch_add(p, v, __ATOMIC_RELAXED, __HIP_MEMORY_SCOPE_AGENT);
}

// ---------------------------------------------------------------------------
// counts per (dst, rel) composite segment — needed for mean normalization
__global__ void count_kernel(const int* __restrict__ dst, const int* __restrict__ rel,
                             int* __restrict__ cnt, int E) {
    int e = blockIdx.x * blockDim.x + threadIdx.x;
    if (e >= E) return;
    atomicAdd(&cnt[dst[e] * RREL + rel[e]], 1);
}

// ---------------------------------------------------------------------------
// One wave per edge. Edges whose relation != r exit immediately (uniform branch).
// 32 lanes x float4 = 128 floats: agg[dst] += norm * x[src]
__global__ void scatter_kernel(const float* __restrict__ x, const int* __restrict__ src,
                               const int* __restrict__ dst, const int* __restrict__ rel,
                               const int* __restrict__ cnt, float* __restrict__ agg,
                               int r, int E) {
    int wid  = (blockIdx.x * blockDim.x + threadIdx.x) >> 5;   // uniform per wave
    int lane = threadIdx.x & 31;
    if (wid >= E) return;
    if (rel[wid] != r) return;
    int s = src[wid], d = dst[wid];
    float norm = 1.0f / fmaxf((float)cnt[d * RREL + r], 1.0f);
    const float4* xs = (const float4*)(x + (size_t)s * HIDD);
    float4 v = xs[lane];
    float* ag = agg + (size_t)d * HIDD + lane * 4;
    atomic_add_f32(ag + 0, v.x * norm);
    atomic_add_f32(ag + 1, v.y * norm);
    atomic_add_f32(ag + 2, v.z * norm);
    atomic_add_f32(ag + 3, v.w * norm);
}

// ---------------------------------------------------------------------------
// out[M,128] = A[M,128] @ W[128,128]  (+ bias-broadcast  OR  + out)   fp32 WMMA.
// 256-thread block = 8 waves; W staged transposed into LDS once per block.
// Each wave computes a 16(M) x 128(N) strip: 8 v8f accumulators, K-loop of 32
// V_WMMA_F32_16X16X4_F32 steps per column tile, B operands are ds_load_b64.
__global__ __launch_bounds__(256) void gemm_wmma(
        const float* __restrict__ A, const float* __restrict__ W,
        float* __restrict__ out, const float* __restrict__ bias,
        int accumulate, int M) {
    extern __shared__ float sWT[];              // [128 cols][WSTR words] = 67.6 KB
    const int tid = threadIdx.x;

    // cooperative transposed W load: thread t reads half of row (t>>1) coalesced,
    // writes column-major into LDS.
    {
        int row = tid >> 1;
        int c0  = (tid & 1) * 64;
        const float4* wr = (const float4*)(W + (size_t)row * HIDD + c0);
#pragma unroll
        for (int i = 0; i < 16; ++i) {
            float4 w4 = wr[i];
            int c = c0 + i * 4;
            sWT[(c + 0) * WSTR + row] = w4.x;
            sWT[(c + 1) * WSTR + row] = w4.y;
            sWT[(c + 2) * WSTR + row] = w4.z;
            sWT[(c + 3) * WSTR + row] = w4.w;
        }
    }
    __syncthreads();

    const int wave   = (blockIdx.x * blockDim.x + tid) >> 5;
    const int strips = M >> 4;
    const int lane = tid & 31;
    const int half = lane >> 4;                 // K-split / M-split selector
    const int lm   = lane & 15;

    if (wave < strips) {                        // tail waves already passed the barrier
        const int m0 = wave << 4;

        v8f acc[8];
        if (accumulate) {
#pragma unroll
            for (int nt = 0; nt < 8; ++nt)
#pragma unroll
                for (int v = 0; v < 8; ++v)
                    acc[nt][v] = out[(size_t)(m0 + v + 8 * half) * HIDD + nt * 16 + lm];
        } else {
#pragma unroll
            for (int nt = 0; nt < 8; ++nt) {
                float bv = bias ? bias[nt * 16 + lm] : 0.0f;
#pragma unroll
                for (int v = 0; v < 8; ++v) acc[nt][v] = bv;
            }
        }

        const float* arow = A + (size_t)(m0 + lm) * HIDD;
        const float* bcol = sWT + lm * WSTR;    // + nt*16*WSTR + ka
        for (int k0 = 0; k0 < HIDD; k0 += 4) {
            int ka = k0 + 2 * half;             // even -> 8B-aligned LDS b64 reads
            v2f a;
            a.x = arow[ka];
            a.y = arow[ka + 1];
#pragma unroll
            for (int nt = 0; nt < 8; ++nt) {
                v2f b = *(const v2f*)(bcol + nt * 16 * WSTR + ka);
                acc[nt] = __builtin_amdgcn_wmma_f32_16x16x4_f32(
                    false, a, false, b, (short)0, acc[nt], false, false);
            }
        }

#pragma unroll
        for (int nt = 0; nt < 8; ++nt)
#pragma unroll
            for (int v = 0; v < 8; ++v)
                out[(size_t)(m0 + v + 8 * half) * HIDD + nt * 16 + lm] = acc[nt][v];
    }
}

// ---------------------------------------------------------------------------
__global__ void relu_kernel(float* __restrict__ h, int n4) {
    int i = blockIdx.x * blockDim.x + threadIdx.x;
    if (i >= n4) return;
    float4* p = (float4*)h;
    float4 v = p[i];
    v.x = fmaxf(v.x, 0.0f);
    v.y = fmaxf(v.y, 0.0f);
    v.z = fmaxf(v.z, 0.0f);
    v.w = fmaxf(v.w, 0.0f);
    p[i] = v;
}

// One wave per node: pooled[batch[n]] += h[n]; gcnt[batch[n]] += 1
__global__ void pool_kernel(const float* __restrict__ h, const int* __restrict__ batch,
                            float* __restrict__ pooled, float* __restrict__ gcnt, int N) {
    int wid  = (blockIdx.x * blockDim.x + threadIdx.x) >> 5;
    int lane = threadIdx.x & 31;
    if (wid >= N) return;
    int g = batch[wid];
    const float4* hs = (const float4*)(h + (size_t)wid * HIDD);
    float4 v = hs[lane];
    float* pg = pooled + (size_t)g * HIDD + lane * 4;
    atomic_add_f32(pg + 0, v.x);
    atomic_add_f32(pg + 1, v.y);
    atomic_add_f32(pg + 2, v.z);
    atomic_add_f32(pg + 3, v.w);
    if (lane == 0) atomic_add_f32(&gcnt[g], 1.0f);
}

// 64 graphs x 16 classes = 1024 threads; out = (pooled/cnt) @ linW + linb
__global__ void final_kernel(const float* __restrict__ pooled, const float* __restrict__ gcnt,
                             const float* __restrict__ linW, const float* __restrict__ linb,
                             float* __restrict__ out) {
    int t = threadIdx.x;
    int g = t >> 4, c = t & 15;
    float inv = 1.0f / fmaxf(gcnt[g], 1.0f);
    float acc = 0.0f;
    for (int h = 0; h < HIDD; ++h)
        acc += pooled[g * HIDD + h] * linW[h * NCLS + c];
    out[g * NCLS + c] = acc * inv + linb[c];
}

// ---------------------------------------------------------------------------
extern "C" void kernel_launch(void* const* d_in, const int* in_sizes, int n_in,
                              void* d_out, int out_size, void* d_ws, size_t ws_size,
                              hipStream_t stream) {
    const float* x     = (const float*)d_in[0];
    const int*   eidx  = (const int*)d_in[1];
    const int*   etype = (const int*)d_in[2];
    const int*   batch = (const int*)d_in[3];
    const float* W1    = (const float*)d_in[4];
    const float* root1 = (const float*)d_in[5];
    const float* b1    = (const float*)d_in[6];
    const float* W2    = (const float*)d_in[7];
    const float* root2 = (const float*)d_in[8];
    const float* b2    = (const float*)d_in[9];
    const float* linW  = (const float*)d_in[10];
    const float* linb  = (const float*)d_in[11];

    const int E = in_sizes[2];          // edge_type length
    const int N = in_sizes[3];          // batch length
    const int* src = eidx;              // edge_index[0]
    const int* dst = eidx + E;          // edge_index[1]

    // workspace carve-up (~80 MB)
    char* ws = (char*)d_ws;
    int*   cnt    = (int*)ws;    ws += (size_t)N * RREL * sizeof(int);
    float* agg    = (float*)ws;  ws += (size_t)N * HIDD * sizeof(float);
    float* h1     = (float*)ws;  ws += (size_t)N * HIDD * sizeof(float);
    float* h2     = (float*)ws;  ws += (size_t)N * HIDD * sizeof(float);
    float* pooled = (float*)ws;  ws += (size_t)NGRAPH * HIDD * sizeof(float);
    float* gcnt   = (float*)ws;  ws += (size_t)NGRAPH * sizeof(float);

    const int strips      = N / 16;
    const int gemm_blocks = (strips + 7) / 8;                 // 8 waves / 256-thread block
    const size_t gemm_lds = (size_t)HIDD * WSTR * sizeof(float);  // 67.6 KB
    const int scat_blocks = (int)(((size_t)E * 32 + 255) / 256);
    const int relu_blocks = (N * (HIDD / 4) + 255) / 256;

    // (dst, rel) degree counts
    hipMemsetAsync(cnt, 0, (size_t)N * RREL * sizeof(int), stream);
    count_kernel<<<(E + 255) / 256, 256, 0, stream>>>(dst, etype, cnt, E);

    // ---- layer 1: h1 = relu( sum_r agg_r @ W1_r + x @ root1 + b1 )
    gemm_wmma<<<gemm_blocks, 256, gemm_lds, stream>>>(x, root1, h1, b1, 0, N);
    for (int r = 0; r < RREL; ++r) {
        hipMemsetAsync(agg, 0, (size_t)N * HIDD * sizeof(float), stream);
        scatter_kernel<<<scat_blocks, 256, 0, stream>>>(x, src, dst, etype, cnt, agg, r, E);
        gemm_wmma<<<gemm_blocks, 256, gemm_lds, stream>>>(agg, W1 + (size_t)r * HIDD * HIDD,
                                                          h1, nullptr, 1, N);
    }
    relu_kernel<<<relu_blocks, 256, 0, stream>>>(h1, N * (HIDD / 4));

    // ---- layer 2: h2 = relu( sum_r agg_r @ W2_r + h1 @ root2 + b2 )
    gemm_wmma<<<gemm_blocks, 256, gemm_lds, stream>>>(h1, root2, h2, b2, 0, N);
    for (int r = 0; r < RREL; ++r) {
        hipMemsetAsync(agg, 0, (size_t)N * HIDD * sizeof(float), stream);
        scatter_kernel<<<scat_blocks, 256, 0, stream>>>(h1, src, dst, etype, cnt, agg, r, E);
        gemm_wmma<<<gemm_blocks, 256, gemm_lds, stream>>>(agg, W2 + (size_t)r * HIDD * HIDD,
                                                          h2, nullptr, 1, N);
    }
    relu_kernel<<<relu_blocks, 256, 0, stream>>>(h2, N * (HIDD / 4));

    // ---- pooling + classifier head
    hipMemsetAsync(pooled, 0, (size_t)NGRAPH * HIDD * sizeof(float), stream);
    hipMemsetAsync(gcnt, 0, (size_t)NGRAPH * sizeof(float), stream);
    pool_kernel<<<(int)(((size_t)N * 32 + 255) / 256), 256, 0, stream>>>(h2, batch, pooled, gcnt, N);
    final_kernel<<<1, NGRAPH * NCLS, 0, stream>>>(pooled, gcnt, linW, linb, (float*)d_out);
}